// CategoricalGINEncoder_32280974197455
// MI455X (gfx1250) — compile-verified
//
#include <hip/hip_runtime.h>

typedef __attribute__((ext_vector_type(2))) float v2f;
typedef __attribute__((ext_vector_type(8))) float v8f;

#define N_NODES  100000
#define N_FIELDS 4
#define N_CAT    1000
#define EMB      32
#define IN_DIM   128   // = HID

// ---------------------------------------------------------------------------
// 1) Embedding gather: X[n, f*32+e] = tables[f][x_cat[n,f]][e]
// ---------------------------------------------------------------------------
__global__ __launch_bounds__(256)
void embed_kernel(const int* __restrict__ x_cat,
                  const float* __restrict__ tables,
                  float* __restrict__ X) {
  int t = blockIdx.x * blockDim.x + threadIdx.x;      // n*128 + j
  if (t >= N_NODES * IN_DIM) return;
  int n = t >> 7;
  int j = t & 127;
  int f = j >> 5;
  int e = j & 31;
  int idx = x_cat[n * N_FIELDS + f];
  X[t] = tables[((size_t)f * N_CAT + (size_t)idx) * EMB + e];
}

// ---------------------------------------------------------------------------
// 2) Zero-fill (float4 vectorized)
// ---------------------------------------------------------------------------
__global__ __launch_bounds__(256)
void zero_kernel(float4* __restrict__ p, int n4) {
  int t = blockIdx.x * blockDim.x + threadIdx.x;
  if (t < n4) p[t] = make_float4(0.f, 0.f, 0.f, 0.f);
}

// ---------------------------------------------------------------------------
// 3) Edge scatter: AGG[dst] += X[src]   (dim = 128; one thread per edge x
//    4-float chunk; float4 gather, 4x global_atomic_add_f32 — AGG is
//    L2-resident (51MB << 192MB L2) so atomics run at L2 rate)
// ---------------------------------------------------------------------------
__global__ __launch_bounds__(256)
void scatter_kernel(const float* __restrict__ X,
                    const int* __restrict__ src,
                    const int* __restrict__ dst,
                    float* __restrict__ AGG,
                    int n_edges) {
  int t = blockIdx.x * blockDim.x + threadIdx.x;
  int e = t >> 5;                       // 32 chunks of 4 floats = 128 dims
  if (e >= n_edges) return;
  int c = (t & 31) << 2;
  int s = src[e];
  int d = dst[e];
  const float4 v = *(const float4*)(X + (size_t)s * 128 + c);
  float* o = AGG + (size_t)d * 128 + c;
  atomicAdd(o + 0, v.x);
  atomicAdd(o + 1, v.y);
  atomicAdd(o + 2, v.z);
  atomicAdd(o + 3, v.w);
}

// ---------------------------------------------------------------------------
// 4) Fused GIN-MLP GEMM using V_WMMA_F32_16X16X4_F32 (full fp32 path).
//    C[M,NOUT] = act( (A (+AGG)) @ W[K,NOUT] + bias )
//    Wave = one 16x16 output tile. K-loop of K/4 WMMAs, fully unrolled.
//    A frag (16x4 f32): lanes 0-15 hold K=k0,k0+1; lanes 16-31 hold k0+2,k0+3
//    B frag (4x16 f32): same K split, N striped over lane%16
//    C/D: 8 VGPRs; lanes 0-15 -> M=r, lanes 16-31 -> M=8+r; N = lane%16
// ---------------------------------------------------------------------------
template<int K, int NOUT, bool RELU, bool HAS_AGG>
__global__ __launch_bounds__(256)
void gin_gemm_wmma(const float* __restrict__ A,
                   const float* __restrict__ AGG,
                   const float* __restrict__ W,
                   const float* __restrict__ bias,
                   float* __restrict__ C) {
  const int lane = threadIdx.x;        // 0..31 (wave32)
  const int half = lane >> 4;          // 0 or 1
  const int l16  = lane & 15;
  const int m0   = blockIdx.x * 16;
  const int n0   = threadIdx.y * 16;   // one wave per 16-col slab

  const float* arow = A + (size_t)(m0 + l16) * K + 2 * half;
  const float* grow = HAS_AGG ? (AGG + (size_t)(m0 + l16) * K + 2 * half)
                              : nullptr;
  const float* wcol = W + (size_t)(2 * half) * NOUT + n0 + l16;

  v8f acc = {0.f, 0.f, 0.f, 0.f, 0.f, 0.f, 0.f, 0.f};

#pragma unroll
  for (int kk = 0; kk < K / 4; ++kk) {
    v2f a = *(const v2f*)(arow + 4 * kk);          // 8B-aligned pair load
    if (HAS_AGG) {
      v2f g = *(const v2f*)(grow + 4 * kk);
      a.x += g.x;
      a.y += g.y;
    }
    v2f b;
    b.x = wcol[(size_t)(4 * kk + 0) * NOUT];
    b.y = wcol[(size_t)(4 * kk + 1) * NOUT];
    // D = A(16x4) x B(4x16) + C   -> v_wmma_f32_16x16x4_f32
    acc = __builtin_amdgcn_wmma_f32_16x16x4_f32(
        /*neg_a=*/false, a, /*neg_b=*/false, b,
        /*c_mod=*/(short)0, acc, /*reuse_a=*/false, /*reuse_b=*/false);
  }

  const int col = n0 + l16;
  const float bv = bias[col];
  float* crow = C + (size_t)(m0 + 8 * half) * NOUT + col;
#pragma unroll
  for (int r = 0; r < 8; ++r) {
    float v = acc[r] + bv;
    if (RELU) v = fmaxf(v, 0.f);
    crow[(size_t)r * NOUT] = v;
  }
}

// ---------------------------------------------------------------------------
// Host-side orchestration
// ---------------------------------------------------------------------------
extern "C" void kernel_launch(void* const* d_in, const int* in_sizes, int n_in,
                              void* d_out, int out_size, void* d_ws, size_t ws_size,
                              hipStream_t stream) {
  const int*   x_cat  = (const int*)d_in[0];
  const int*   edge   = (const int*)d_in[1];
  const float* tables = (const float*)d_in[2];
  const float* w1a    = (const float*)d_in[3];
  const float* b1a    = (const float*)d_in[4];
  const float* w1b    = (const float*)d_in[5];
  const float* b1b    = (const float*)d_in[6];
  const float* w2a    = (const float*)d_in[7];
  const float* b2a    = (const float*)d_in[8];
  const float* w2b    = (const float*)d_in[9];
  const float* b2b    = (const float*)d_in[10];
  float* out = (float*)d_out;

  const int n_edges = in_sizes[1] / 2;
  const int* src = edge;
  const int* dst = edge + n_edges;

  // Workspace: 3 buffers of N*128 floats (X, AGG, H) = 153.6 MB
  float* X   = (float*)d_ws;
  float* AGG = X   + (size_t)N_NODES * 128;
  float* H   = AGG + (size_t)N_NODES * 128;

  const int nx = N_NODES * 128;                 // 12.8M floats
  const int zero_blocks = (nx / 4 + 255) / 256;
  const long long sc_threads = (long long)n_edges * 32;
  const int sc_blocks = (int)((sc_threads + 255) / 256);
  const dim3 blk8(32, 8);                       // 8 waves: 16x128 tile/block
  const dim3 blk4(32, 4);                       // 4 waves: 16x64 tile/block
  const int mtiles = N_NODES / 16;              // 6250 (exact)

  // --- embedding gather ---
  embed_kernel<<<(nx + 255) / 256, 256, 0, stream>>>(x_cat, tables, X);

  // --- conv1 aggregation ---
  zero_kernel<<<zero_blocks, 256, 0, stream>>>((float4*)AGG, nx / 4);
  scatter_kernel<<<sc_blocks, 256, 0, stream>>>(X, src, dst, AGG, n_edges);

  // --- conv1 MLP:  H = relu((X+AGG)@w1a + b1a);  X = relu(H@w1b + b1b) ---
  gin_gemm_wmma<128, 128, true,  true ><<<mtiles, blk8, 0, stream>>>(X, AGG, w1a, b1a, H);
  gin_gemm_wmma<128, 128, true,  false><<<mtiles, blk8, 0, stream>>>(H, nullptr, w1b, b1b, X);

  // --- conv2 aggregation ---
  zero_kernel<<<zero_blocks, 256, 0, stream>>>((float4*)AGG, nx / 4);
  scatter_kernel<<<sc_blocks, 256, 0, stream>>>(X, src, dst, AGG, n_edges);

  // --- conv2 MLP:  H = relu((X+AGG)@w2a + b2a);  out = H@w2b + b2b ---
  gin_gemm_wmma<128, 64,  true,  true ><<<mtiles, blk4, 0, stream>>>(X, AGG, w2a, b2a, H);
  gin_gemm_wmma<64,  64,  false, false><<<mtiles, blk4, 0, stream>>>(H, nullptr, w2b, b2b, out);
}